// TrivecRenderer_47425028883023
// MI455X (gfx1250) — compile-verified
//
#include <hip/hip_runtime.h>
#include <math.h>

// ---------------------------------------------------------------------------
// TrivecRenderer for MI455X (gfx1250, wave32).
// Pipeline:
//   k_init_rays      : init per-ray scratch (rays with no samples -> bg only)
//   k_sample         : fused per-sample gather + feature/sigma/color/alpha
//   k_scan_block     : 1024-elem block scans; each wave scans 256 elems via
//                      V_WMMA_F32_16X16X4_F32 matmul-scan (X * U_triangular)
//   k_scan_partials  : single-block Hillis-Steele scan of 977 block totals
//   k_finalize_prefix: global inclusive prefix + ray segment boundaries
//   k_rays           : per-ray serial accumulation -> rgb/depth/alpha maps
// ---------------------------------------------------------------------------

typedef __attribute__((ext_vector_type(2))) float v2f;
typedef __attribute__((ext_vector_type(8))) float v8f;

#define NS        1000000      // N_SAMPLES
#define NR        100000       // N_RAYS (250*400)
#define C_CH      16
#define DIMV      8
#define STEP_SZ   0.005f
#define DSCALE    1.0f         // min(1/(1-0), 25) with DENSITY_SHIFT = 0
#define SCANBLK   1024

__device__ __forceinline__ float softplusf(float x) {
    // matches jax.nn.softplus = max(x,0) + log1p(exp(-|x|))
    return fmaxf(x, 0.0f) + log1pf(expf(-fabsf(x)));
}
__device__ __forceinline__ float sigmoidf_(float x) {
    return 1.0f / (1.0f + expf(-x));
}

// --------------------------- per-sample pass -------------------------------
__global__ __launch_bounds__(256) void k_sample(
    const float* __restrict__ trivecs, const float* __restrict__ dens,
    const float* __restrict__ colors,  const float* __restrict__ gws,
    const float* __restrict__ gwl,     const float* __restrict__ tmin,
    const int* __restrict__ gis,       const int* __restrict__ gil,
    const int* __restrict__ tid_arr,   const int* __restrict__ ray_arr,
    const int* __restrict__ step_arr,
    float* __restrict__ logt, float* __restrict__ alpha,
    float* __restrict__ zarr, float* __restrict__ rgbp)
{
    int i = blockIdx.x * 256 + threadIdx.x;
    if (i >= NS) return;

    int tid = tid_arr[i];
    int is0 = gis[3*i+0], is1 = gis[3*i+1], is2 = gis[3*i+2];
    int il0 = gil[3*i+0], il1 = gil[3*i+1], il2 = gil[3*i+2];
    float ws0 = gws[3*i+0], ws1 = gws[3*i+1], ws2 = gws[3*i+2];
    float wl0 = gwl[3*i+0], wl1 = gwl[3*i+1], wl2 = gwl[3*i+2];

    const float* base = trivecs + (long)tid * (C_CH * 3 * DIMV);
    __builtin_prefetch(base, 0, 1);        // global_prefetch_b8: pull tensoRF block
    const float* dp = dens   + (long)tid * C_CH;
    const float* cp = colors + (long)tid * C_CH * 3;

    float sigma = 0.0f, c0 = 0.0f, c1 = 0.0f, c2 = 0.0f;
#pragma unroll
    for (int c = 0; c < C_CH; ++c) {
        const float* p = base + c * (3 * DIMV);
        float f0 = p[       is0] * ws0 + p[       il0] * wl0;
        float f1 = p[DIMV + is1] * ws1 + p[DIMV + il1] * wl1;
        float f2 = p[2*DIMV+is2] * ws2 + p[2*DIMV+il2] * wl2;
        float f  = f0 * f1 * f2;
        sigma = fmaf(f, dp[c],     sigma);
        c0    = fmaf(f, cp[3*c+0], c0);
        c1    = fmaf(f, cp[3*c+1], c1);
        c2    = fmaf(f, cp[3*c+2], c2);
    }
    float density = softplusf(sigma) * DSCALE;
    float lt = -density * STEP_SZ;
    logt[i]  = lt;
    alpha[i] = 1.0f - expf(lt);
    rgbp[i]        = sigmoidf_(c0);
    rgbp[NS + i]   = sigmoidf_(c1);
    rgbp[2*NS + i] = sigmoidf_(c2);
    int r = ray_arr[i];
    zarr[i] = tmin[r] + (float)step_arr[i] * STEP_SZ;
}

// ----------------- block scan: WMMA matmul-scan per wave -------------------
// Wave32 scans 256 contiguous floats viewed as a 16x16 matrix X (elem = 16M+N).
// Row-inclusive scan = X * U  (U upper-triangular ones), computed as a chain
// of 4x V_WMMA_F32_16X16X4_F32 (K-slices of 4). Cross-row offsets from the
// last column of the result via a tiny LDS scan.
__global__ __launch_bounds__(256) void k_scan_block(
    const float* __restrict__ logt, float* __restrict__ prefix,
    float* __restrict__ partials, int n)
{
    __shared__ float tile[SCANBLK];
    __shared__ float wrow[8][16];
    __shared__ float wtot[8];
    __shared__ float woff[8];

    int  t    = threadIdx.x;
    int  blk  = blockIdx.x;
    long gb   = (long)blk * SCANBLK;

#pragma unroll
    for (int j = 0; j < 4; ++j) {
        long g = gb + t * 4 + j;
        tile[t * 4 + j] = (g < n) ? logt[g] : 0.0f;
    }
    __syncthreads();

    int lane  = t & 31;
    int wv    = t >> 5;         // 8 waves per block
    int half  = lane >> 4;      // lane half: 0 or 1
    int m     = lane & 15;      // A: row M    ; B/D: col N
    int wbase = wv * 256;

    v8f acc = {0.f, 0.f, 0.f, 0.f, 0.f, 0.f, 0.f, 0.f};
#pragma unroll
    for (int ks = 0; ks < 4; ++ks) {
        // A layout (16x4 f32): VGPR v, lane -> M = lane%16, K = v + 2*(lane/16)
        int k0 = 4 * ks + 2 * half;
        v2f a, b;
        a.x = tile[wbase + 16 * m + k0 + 0];
        a.y = tile[wbase + 16 * m + k0 + 1];
        // B = U slice: B[K][N] = (K <= N); same (v, half) -> K mapping, N = lane%16
        b.x = (k0 + 0 <= m) ? 1.0f : 0.0f;
        b.y = (k0 + 1 <= m) ? 1.0f : 0.0f;
        acc = __builtin_amdgcn_wmma_f32_16x16x4_f32(
            /*neg_a=*/false, a, /*neg_b=*/false, b,
            /*c_mod=*/(short)0, acc, /*reuse_a=*/false, /*reuse_b=*/false);
    }

    // D layout: component r holds (M = r + 8*half, N = m). Row totals live in N==15.
    if (m == 15) {
#pragma unroll
        for (int r = 0; r < 8; ++r) wrow[wv][r + 8 * half] = acc[r];
    }
    __syncthreads();
    if (lane == 0) {
        float s = 0.0f;
#pragma unroll
        for (int q = 0; q < 16; ++q) { float v = wrow[wv][q]; wrow[wv][q] = s; s += v; }
        wtot[wv] = s;
    }
    __syncthreads();
#pragma unroll
    for (int r = 0; r < 8; ++r) {
        int M = r + 8 * half;
        tile[wbase + 16 * M + m] = acc[r] + wrow[wv][M];   // wave-local inclusive
    }
    __syncthreads();
    if (t == 0) {
        float s = 0.0f;
#pragma unroll
        for (int q = 0; q < 8; ++q) { woff[q] = s; s += wtot[q]; }
        partials[blk] = s;
    }
    __syncthreads();
#pragma unroll
    for (int j = 0; j < 4; ++j) {
        int  idx = t * 4 + j;
        long g   = gb + idx;
        if (g < n) prefix[g] = tile[idx] + woff[idx >> 8];  // block-local inclusive
    }
}

// ------------------- single-block scan of block partials -------------------
__global__ __launch_bounds__(1024) void k_scan_partials(float* partials, int nb)
{
    __shared__ float s[1024];
    int t = threadIdx.x;
    s[t] = (t < nb) ? partials[t] : 0.0f;
    __syncthreads();
    for (int off = 1; off < 1024; off <<= 1) {
        float add = (t >= off) ? s[t - off] : 0.0f;
        __syncthreads();
        s[t] += add;
        __syncthreads();
    }
    if (t < nb) partials[t] = s[t];   // inclusive scan of block totals
}

// --------------- global prefix + ray segment boundary detection ------------
__global__ __launch_bounds__(256) void k_finalize_prefix(
    const float* __restrict__ logt, float* __restrict__ prefix,
    const float* __restrict__ partials, const int* __restrict__ ray_arr,
    float* __restrict__ ray_base, float* __restrict__ ray_lastI,
    int* __restrict__ ray_start, int n)
{
    int i = blockIdx.x * 256 + threadIdx.x;
    if (i >= n) return;
    int   blk = i >> 10;
    float I   = prefix[i] + ((blk > 0) ? partials[blk - 1] : 0.0f);
    prefix[i] = I;                                   // global inclusive cumsum
    int r = ray_arr[i];
    if (i == 0     || ray_arr[i - 1] != r) { ray_base[r] = I - logt[i]; ray_start[r] = i; }
    if (i == n - 1 || ray_arr[i + 1] != r) { ray_lastI[r] = I; }
}

__global__ __launch_bounds__(256) void k_init_rays(
    float* __restrict__ ray_base, float* __restrict__ ray_lastI,
    int* __restrict__ ray_start)
{
    int r = blockIdx.x * 256 + threadIdx.x;
    if (r < NR) { ray_base[r] = 0.0f; ray_lastI[r] = 0.0f; ray_start[r] = -1; }
}

// --------------------------- per-ray accumulation --------------------------
__global__ __launch_bounds__(256) void k_rays(
    const float* __restrict__ logt,  const float* __restrict__ alpha,
    const float* __restrict__ zarr,  const float* __restrict__ rgbp,
    const float* __restrict__ prefix, const int* __restrict__ ray_arr,
    const float* __restrict__ ray_base, const float* __restrict__ ray_lastI,
    const int* __restrict__ ray_start,  const float* __restrict__ bg,
    float* __restrict__ out)
{
    int r = blockIdx.x * 256 + threadIdx.x;
    if (r >= NR) return;
    float base = ray_base[r];
    float bgw  = expf(ray_lastI[r] - base);          // exp(seg_total)
    float a0 = 0.f, a1 = 0.f, a2 = 0.f, d = 0.f;
    int j = ray_start[r];
    if (j >= 0) {
        while (j < NS && ray_arr[j] == r) {
            float E = prefix[j] - logt[j];           // global exclusive cumsum
            float w = alpha[j] * expf(E - base);     // alpha * exp(excl)
            a0 = fmaf(w, rgbp[j],        a0);
            a1 = fmaf(w, rgbp[NS + j],   a1);
            a2 = fmaf(w, rgbp[2*NS + j], a2);
            d  = fmaf(w, zarr[j],        d);
            ++j;
        }
    }
    out[0 * NR + r] = a0 + bgw * bg[0];              // rgb_map.T -> (3, H*W)
    out[1 * NR + r] = a1 + bgw * bg[1];
    out[2 * NR + r] = a2 + bgw * bg[2];
    out[3 * NR + r] = d;                             // depth_map
    out[4 * NR + r] = 1.0f - bgw;                    // alpha_map
}

// ------------------------------- launcher ----------------------------------
extern "C" void kernel_launch(void* const* d_in, const int* in_sizes, int n_in,
                              void* d_out, int out_size, void* d_ws, size_t ws_size,
                              hipStream_t stream) {
    const float* trivecs = (const float*)d_in[0];
    const float* dens    = (const float*)d_in[1];
    const float* colors  = (const float*)d_in[2];
    const float* gws     = (const float*)d_in[3];
    const float* gwl     = (const float*)d_in[4];
    const float* tmin    = (const float*)d_in[5];
    const float* bg      = (const float*)d_in[6];
    const int*   gis     = (const int*)d_in[7];
    const int*   gil     = (const int*)d_in[8];
    const int*   tid     = (const int*)d_in[9];
    const int*   ray     = (const int*)d_in[10];
    const int*   step    = (const int*)d_in[11];

    // workspace layout (floats): ~29.3 MB total
    float* ws       = (float*)d_ws;
    float* logt     = ws;                 // NS
    float* alpha    = ws + (size_t)NS;    // NS
    float* zarr     = ws + 2*(size_t)NS;  // NS
    float* rgbp     = ws + 3*(size_t)NS;  // 3*NS (3 planes)
    float* prefix   = ws + 6*(size_t)NS;  // NS
    float* partials = ws + 7*(size_t)NS;  // <=1024
    float* rbase    = partials + 1024;    // NR
    float* rlast    = rbase + NR;         // NR
    int*   rstart   = (int*)(rlast + NR); // NR

    const int NB = (NS + SCANBLK - 1) / SCANBLK;   // 977 <= 1024

    k_init_rays<<<(NR + 255) / 256, 256, 0, stream>>>(rbase, rlast, rstart);
    k_sample<<<(NS + 255) / 256, 256, 0, stream>>>(
        trivecs, dens, colors, gws, gwl, tmin, gis, gil, tid, ray, step,
        logt, alpha, zarr, rgbp);
    k_scan_block<<<NB, 256, 0, stream>>>(logt, prefix, partials, NS);
    k_scan_partials<<<1, 1024, 0, stream>>>(partials, NB);
    k_finalize_prefix<<<(NS + 255) / 256, 256, 0, stream>>>(
        logt, prefix, partials, ray, rbase, rlast, rstart, NS);
    k_rays<<<(NR + 255) / 256, 256, 0, stream>>>(
        logt, alpha, zarr, rgbp, prefix, ray, rbase, rlast, rstart, bg,
        (float*)d_out);
}